// MessagePassing_32847909879996
// MI455X (gfx1250) — compile-verified
//
#include <hip/hip_runtime.h>

#define D_FEAT 128
#define EDGES_PER_WAVE 32

// ---------------------------------------------------------------------------
// Zero the output accumulator (harness poisons d_out with 0xAA before timing).
// ---------------------------------------------------------------------------
__global__ void mp_zero_kernel(float4* __restrict__ out, int n4) {
    int i = blockIdx.x * blockDim.x + threadIdx.x;
    if (i < n4) out[i] = make_float4(0.f, 0.f, 0.f, 0.f);
}

// ---------------------------------------------------------------------------
// Edge-parallel gather + scatter-add (out[dst] += x[src], rows of 128 f32).
//
// One wave32 owns a tile of 32 edges:
//   - each lane loads the low dword of one edge's (dst, src) int64
//     (node ids < 2^31, so the high dword is zero),
//   - each lane prefetches all four 128B lines of its edge's source row
//     (gfx1250 global_prefetch_b8) before the serial walk,
//   - the wave then walks the 32 edges: v_readlane_b32 broadcasts the ids
//     to SGPRs, every lane does one global_load_b128 (16B of the 512B row)
//     and four no-return global_atomic_add_f32 into the destination row.
//
// The gather is software-pipelined: edge j+1's row load is issued before
// edge j's atomics, so one full row-load latency always overlaps atomic
// issue. x (25.6MB), out (25.6MB) and edge_index (25.6MB) all fit in the
// 192MB L2, so steady-state traffic is L2-resident; the kernel is bounded
// by L2 gather bandwidth and L2 f32-atomic throughput, not HBM.
// ---------------------------------------------------------------------------
__global__ __launch_bounds__(256)
void mp_scatter_add_kernel(const float* __restrict__ x,
                           const int* __restrict__ dst_lo,   // edge_index row 0, viewed as i32 pairs
                           const int* __restrict__ src_lo,   // edge_index row 1, viewed as i32 pairs
                           float* __restrict__ out,
                           int n_edges) {
    const int lane = threadIdx.x & 31;
    const int wave = blockIdx.x * (blockDim.x >> 5) + (threadIdx.x >> 5);
    const int base = wave * EDGES_PER_WAVE;
    if (base >= n_edges) return;            // wave-uniform exit

    const int e = base + lane;
    int d = 0, s = 0;
    if (e < n_edges) {
        d = dst_lo[2 * e];                  // low dword of int64 index (value < 50000)
        s = src_lo[2 * e];
        // Warm the full 512B source row (4 x 128B lines) this wave will
        // gather shortly. Lowers to gfx1250 global_prefetch_b8 (per-lane).
        const float* row = x + (long long)s * D_FEAT;
        __builtin_prefetch(row +  0, 0, 3);
        __builtin_prefetch(row + 32, 0, 3);
        __builtin_prefetch(row + 64, 0, 3);
        __builtin_prefetch(row + 96, 0, 3);
    }

    const int tile_n = min(EDGES_PER_WAVE, n_edges - base);

    // Prologue: issue edge 0's row load.
    int sj = __builtin_amdgcn_readlane(s, 0);
    int dj = __builtin_amdgcn_readlane(d, 0);
    float4 v = ((const float4*)(x + (long long)sj * D_FEAT))[lane];

    for (int j = 0; j < tile_n; ++j) {
        // Issue edge j+1's broadcast + row load before edge j's atomics.
        float4 vnext;
        int dnext = dj;
        if (j + 1 < tile_n) {
            const int sn = __builtin_amdgcn_readlane(s, j + 1);
            dnext        = __builtin_amdgcn_readlane(d, j + 1);
            vnext = ((const float4*)(x + (long long)sn * D_FEAT))[lane];
        }

        // Four hardware f32 atomics (global_atomic_add_f32, no-return form).
        float* o = out + (long long)dj * D_FEAT + (lane << 2);
        unsafeAtomicAdd(o + 0, v.x);
        unsafeAtomicAdd(o + 1, v.y);
        unsafeAtomicAdd(o + 2, v.z);
        unsafeAtomicAdd(o + 3, v.w);

        v  = vnext;
        dj = dnext;
    }
}

extern "C" void kernel_launch(void* const* d_in, const int* in_sizes, int n_in,
                              void* d_out, int out_size, void* d_ws, size_t ws_size,
                              hipStream_t stream) {
    (void)n_in; (void)d_ws; (void)ws_size;

    const float*     x          = (const float*)d_in[0];
    const long long* edge_index = (const long long*)d_in[1];
    const int        n_edges    = in_sizes[1] / 2;       // [2, E] int64
    float*           out        = (float*)d_out;

    // 1) Zero the accumulator (out_size = 50000*128, divisible by 4).
    const int n4 = out_size / 4;
    mp_zero_kernel<<<(n4 + 255) / 256, 256, 0, stream>>>((float4*)out, n4);

    // 2) Gather + scatter-add. Row 0 = destinations, row 1 = sources.
    const int* dst_lo = (const int*)(edge_index);             // row 0
    const int* src_lo = (const int*)(edge_index + n_edges);   // row 1

    const int waves  = (n_edges + EDGES_PER_WAVE - 1) / EDGES_PER_WAVE;
    const int blocks = (waves + 7) / 8;                       // 256 thr = 8 wave32
    mp_scatter_add_kernel<<<blocks, 256, 0, stream>>>(x, dst_lo, src_lo, out, n_edges);
}